// IoULoss_75213467287779
// MI455X (gfx1250) — compile-verified
//
#include <hip/hip_runtime.h>

typedef __attribute__((ext_vector_type(2))) float v2f;
typedef __attribute__((ext_vector_type(8))) float v8f;

#define IOU_EPS 1e-7f

// Exact-f32 wave32 all-lane sum using V_WMMA_F32_16X16X4_F32 as a reduction
// engine.  Stage 1: A(16x4) holds lane values (lanes 0-15 -> K=0, lanes
// 16-31 -> K=2), B = ones => D[m][*] = s[m] + s[m+16] = t[m].  D's VGPR v
// holds t[v] (lanes 0-15) and t[v+8] (lanes 16-31).  Stage 2: feed D VGPR
// pairs back as A with C-chaining; 4 accumulating WMMAs sum t[0..15].
// Every lane of the result holds the full 32-lane sum.
__device__ __forceinline__ float wave_sum32_wmma(float s) {
  v2f a;    a[0] = s;    a[1] = 0.0f;
  v2f ones; ones[0] = 1.0f; ones[1] = 1.0f;
  v8f zero = {};
  v8f d = __builtin_amdgcn_wmma_f32_16x16x4_f32(
      false, a, false, ones, (short)0, zero, false, false);
  v8f acc = {};
#pragma unroll
  for (int p = 0; p < 4; ++p) {
    v2f ap; ap[0] = d[2 * p]; ap[1] = d[2 * p + 1];
    acc = __builtin_amdgcn_wmma_f32_16x16x4_f32(
        false, ap, false, ones, (short)0, acc, false, false);
  }
  return acc[0];
}

__global__ __launch_bounds__(256)
void iou_loss_partial(const float4* __restrict__ pred,
                      const float4* __restrict__ tgt,
                      float* __restrict__ part, int n) {
  __shared__ float smem[8];
  int i = blockIdx.x * 256 + threadIdx.x;

  float loss = 0.0f;
  if (i < n) {
    float4 p = pred[i];   // cx, cy, w, h  (global_load_b128)
    float4 t = tgt[i];
    // corners, matching the reference's order of operations
    float px1 = p.x - p.z * 0.5f, py1 = p.y - p.w * 0.5f;
    float px2 = p.x + p.z * 0.5f, py2 = p.y + p.w * 0.5f;
    float tx1 = t.x - t.z * 0.5f, ty1 = t.y - t.w * 0.5f;
    float tx2 = t.x + t.z * 0.5f, ty2 = t.y + t.w * 0.5f;

    float ix1 = fmaxf(px1, tx1), iy1 = fmaxf(py1, ty1);
    float ix2 = fminf(px2, tx2), iy2 = fminf(py2, ty2);
    float iw = fmaxf(ix2 - ix1, 0.0f), ih = fmaxf(iy2 - iy1, 0.0f);
    float inter = iw * ih;

    float a1 = (px2 - px1) * (py2 - py1);
    float a2 = (tx2 - tx1) * (ty2 - ty1);
    float uni = a1 + a2 - inter;

    loss = 1.0f - inter / (uni + IOU_EPS);
  }

  // EXEC is all-1s here (reconverged) -> legal to issue WMMA
  float wsum = wave_sum32_wmma(loss);

  int lane = threadIdx.x & 31;
  int wid  = threadIdx.x >> 5;
  if (lane == 0) smem[wid] = wsum;
  __syncthreads();
  if (threadIdx.x == 0) {
    float b = 0.0f;
#pragma unroll
    for (int k = 0; k < 8; ++k) b += smem[k];   // fixed order: deterministic
    part[blockIdx.x] = b;
  }
}

__global__ __launch_bounds__(32)
void iou_loss_final(const float* __restrict__ part, int nb,
                    float* __restrict__ out, float inv_n) {
  int lane = threadIdx.x;
  float v = 0.0f;
  for (int k = lane; k < nb; k += 32) v += part[k];
  float s = wave_sum32_wmma(v);   // all 32 lanes active -> EXEC all-1s
  if (lane == 0) out[0] = s * inv_n;
}

extern "C" void kernel_launch(void* const* d_in, const int* in_sizes, int n_in,
                              void* d_out, int out_size, void* d_ws, size_t ws_size,
                              hipStream_t stream) {
  (void)n_in; (void)out_size; (void)ws_size;
  const float4* pred = (const float4*)d_in[0];   // (N,4) f32
  const float4* tgt  = (const float4*)d_in[1];   // (N,4) f32
  int n  = in_sizes[0] / 4;                      // N = 8192
  int nb = (n + 255) / 256;                      // 32 blocks

  float* part = (float*)d_ws;                    // nb floats of scratch

  iou_loss_partial<<<nb, 256, 0, stream>>>(pred, tgt, part, n);
  iou_loss_final<<<1, 32, 0, stream>>>(part, nb, (float*)d_out,
                                       1.0f / (float)n);
}